// GCNModel_23708219474023
// MI455X (gfx1250) — compile-verified
//
#include <hip/hip_runtime.h>
#include <hip/hip_bf16.h>

typedef __attribute__((ext_vector_type(2))) float v2f;
typedef __attribute__((ext_vector_type(8))) float v8f;

#define HID 128

// ---------------------------------------------------------------- fills
__global__ void fill_value_kernel(float* __restrict__ p, float val, long long n) {
    long long i = (long long)blockIdx.x * blockDim.x + threadIdx.x;
    if (i < n) p[i] = val;
}

// p[v*128 + f] = bias[f]
__global__ void fill_rows_bias_kernel(float* __restrict__ p, const float* __restrict__ bias,
                                      long long total) {
    long long i = (long long)blockIdx.x * blockDim.x + threadIdx.x;
    if (i < total) p[i] = bias[i & (HID - 1)];
}

// ---------------------------------------------------------------- degree
__global__ void deg_accum_kernel(float* __restrict__ deg, const int* __restrict__ col, int E) {
    int e = blockIdx.x * blockDim.x + threadIdx.x;
    if (e < E) atomicAdd(&deg[col[e]], 1.0f);
}

__global__ void rsqrt_inplace_kernel(float* __restrict__ d, int n) {
    int i = blockIdx.x * blockDim.x + threadIdx.x;
    if (i < n) {
        float v = d[i];
        d[i] = v > 0.0f ? rsqrtf(v) : 0.0f;
    }
}

// ---------------------------------------------------------------- layer-1 GEMM (K = n_feat = 11)
__global__ __launch_bounds__(128) void gemm_k11_kernel(const float* __restrict__ x,
                                                       const float* __restrict__ W1,
                                                       float* __restrict__ xw,
                                                       int N, int K) {
    __shared__ float Ws[16 * HID];
    __shared__ float xs[16];
    int t = threadIdx.x;
    for (int i = t; i < K * HID; i += 128) Ws[i] = W1[i];
    __syncthreads();
    for (int v = blockIdx.x; v < N; v += gridDim.x) {
        if (t < K) xs[t] = x[(long long)v * K + t];
        __syncthreads();
        float s = 0.0f;
        #pragma unroll 11
        for (int k = 0; k < K; ++k) s += xs[k] * Ws[k * HID + t];
        xw[(long long)v * HID + t] = s;
        __syncthreads();
    }
}

// ---------------------------------------------------------------- edge scatter (gather + atomic scatter-add)
// one wave per work item; items [0,E) are edges, [E,E+N) are self-loops
__global__ __launch_bounds__(256) void scatter_kernel(const float* __restrict__ xw,
                                                      float* __restrict__ h,
                                                      const float* __restrict__ dis,
                                                      const int* __restrict__ row,
                                                      const int* __restrict__ col,
                                                      int E, int N) {
    long long item = ((long long)blockIdx.x * blockDim.x + threadIdx.x) >> 5;
    int lane = threadIdx.x & 31;
    if (item >= (long long)E + N) return;
    int r, c;
    float norm;
    if (item < E) {
        // lane 0 loads the (wave-uniform) edge endpoints, broadcast to the wave; each lane computes norm
        if (lane == 0) {
            r = row[item];
            c = col[item];
        } else { r = 0; c = 0; }
        r = __shfl(r, 0, 32);
        c = __shfl(c, 0, 32);
        norm = dis[r] * dis[c];
    } else {
        r = c = (int)(item - E);
        float d = dis[r];
        norm = d * d;
    }
    const float4 v = ((const float4*)(xw + (long long)r * HID))[lane];
    float* dst = h + (long long)c * HID + lane * 4;
    atomicAdd(dst + 0, norm * v.x);
    atomicAdd(dst + 1, norm * v.y);
    atomicAdd(dst + 2, norm * v.z);
    atomicAdd(dst + 3, norm * v.w);
}

// ---------------------------------------------------------------- WMMA GEMM: C[M,128] = op(A[M,128]) @ B[128,128]
// fp32 exact via V_WMMA_F32_16X16X4_F32. 8 waves/block, 64x128 tile/block.
// Flags are template params so the epilogue is branch-free straight-line code;
// GUARD=true only for the single tail block when M % 64 != 0.
#define LDSTR 132  // padded LDS row stride (floats) -> conflict-free frag loads

template <bool RELU_IN, bool HAS_BIAS, bool RELU_OUT, bool GUARD>
__global__ __launch_bounds__(256) void wmma_gemm_k128_kernel(const float* __restrict__ A,
                                                             const float* __restrict__ B,
                                                             float* __restrict__ C,
                                                             const float* __restrict__ bias,
                                                             int M, int m0_base) {
    __shared__ __align__(16) float As[64 * LDSTR];    // ~33.8 KB
    __shared__ __align__(16) float Bt[HID * LDSTR];   // ~67.6 KB (B transposed: Bt[n][k])
    const int tid = threadIdx.x;
    const int lane = tid & 31;
    const int wave = tid >> 5;                 // 0..7 -> N slice
    const int m_block = m0_base + blockIdx.x * 64;

    // stage B transposed (coalesced global read, strided LDS write)
    for (int i = tid; i < HID * HID; i += 256) {
        int k = i >> 7, n = i & (HID - 1);
        Bt[n * LDSTR + k] = B[i];
    }
    // stage A tile (optional fused ReLU; row-guarded only in the tail block)
    for (int i = tid; i < 64 * HID; i += 256) {
        int r = i >> 7, c = i & (HID - 1);
        int gr = m_block + r;
        float v = 0.0f;
        if (!GUARD || gr < M) v = A[(long long)gr * HID + c];
        if (RELU_IN) v = fmaxf(v, 0.0f);
        As[r * LDSTR + c] = v;
    }
    __syncthreads();

    const int n0 = wave * 16;
    const int lrow = lane & 15;
    const int koff = (lane >> 4) * 2;          // lanes 16-31 hold K+2,K+3

    v8f acc[4];
    #pragma unroll
    for (int t = 0; t < 4; ++t)
        #pragma unroll
        for (int j = 0; j < 8; ++j) acc[t][j] = 0.0f;

    #pragma unroll 4
    for (int k = 0; k < HID; k += 4) {
        v2f bfrag = *(const v2f*)&Bt[(n0 + lrow) * LDSTR + k + koff];
        #pragma unroll
        for (int t = 0; t < 4; ++t) {
            v2f afrag = *(const v2f*)&As[(t * 16 + lrow) * LDSTR + k + koff];
            acc[t] = __builtin_amdgcn_wmma_f32_16x16x4_f32(
                false, afrag, false, bfrag, (short)0, acc[t], false, false);
        }
    }

    // store C: VGPR j -> (M = base + j [+8 for hi lanes], N = n0 + lane&15)
    const int col = n0 + lrow;
    const float bv = HAS_BIAS ? bias[col] : 0.0f;
    #pragma unroll
    for (int t = 0; t < 4; ++t) {
        int mbase = m_block + t * 16 + ((lane >> 4) ? 8 : 0);
        #pragma unroll
        for (int j = 0; j < 8; ++j) {
            int rrow = mbase + j;
            float v = acc[t][j];
            if (HAS_BIAS) v += bv;
            if (RELU_OUT) v = fmaxf(v, 0.0f);
            if (!GUARD || rrow < M) C[(long long)rrow * HID + col] = v;
        }
    }
}

// ---------------------------------------------------------------- pooling
// one wave per node: pooled[batch[v]] += relu(h[v]); cnt[batch[v]] += 1
__global__ __launch_bounds__(256) void pool_accum_kernel(const float* __restrict__ h,
                                                         const int* __restrict__ batch,
                                                         float* __restrict__ pooled,
                                                         float* __restrict__ cnt, int N) {
    long long item = ((long long)blockIdx.x * blockDim.x + threadIdx.x) >> 5;
    int lane = threadIdx.x & 31;
    if (item >= N) return;
    int g = batch[item];
    const float4 v = ((const float4*)(h + item * HID))[lane];
    float* dst = pooled + (long long)g * HID + lane * 4;
    atomicAdd(dst + 0, fmaxf(v.x, 0.0f));
    atomicAdd(dst + 1, fmaxf(v.y, 0.0f));
    atomicAdd(dst + 2, fmaxf(v.z, 0.0f));
    atomicAdd(dst + 3, fmaxf(v.w, 0.0f));
    if (lane == 0) atomicAdd(&cnt[g], 1.0f);
}

__global__ void pool_scale_kernel(float* __restrict__ pooled, const float* __restrict__ cnt,
                                  int total) {
    int i = blockIdx.x * blockDim.x + threadIdx.x;
    if (i < total) pooled[i] /= fmaxf(cnt[i >> 7], 1.0f);
}

// ---------------------------------------------------------------- final 128 -> 1 projection
__global__ __launch_bounds__(256) void final_dot_kernel(const float* __restrict__ t,
                                                        const float* __restrict__ Wl2,
                                                        const float* __restrict__ bl2,
                                                        float* __restrict__ out, int G) {
    long long g = ((long long)blockIdx.x * blockDim.x + threadIdx.x) >> 5;
    int lane = threadIdx.x & 31;
    if (g >= G) return;
    const float4 tv = ((const float4*)(t + g * HID))[lane];
    const float4 wv = ((const float4*)Wl2)[lane];
    float s = tv.x * wv.x + tv.y * wv.y + tv.z * wv.z + tv.w * wv.w;
    #pragma unroll
    for (int m = 16; m > 0; m >>= 1) s += __shfl_xor(s, m, 32);
    if (lane == 0) out[g] = s + bl2[0];
}

// ----------------------------------------------------------------
extern "C" void kernel_launch(void* const* d_in, const int* in_sizes, int n_in,
                              void* d_out, int out_size, void* d_ws, size_t ws_size,
                              hipStream_t stream) {
    const float* x   = (const float*)d_in[0];
    const float* W1  = (const float*)d_in[1];
    const float* b1  = (const float*)d_in[2];
    const float* W2  = (const float*)d_in[3];
    const float* b2  = (const float*)d_in[4];
    const float* Wl1 = (const float*)d_in[5];
    const float* bl1 = (const float*)d_in[6];
    const float* Wl2 = (const float*)d_in[7];
    const float* bl2 = (const float*)d_in[8];
    const int* edge  = (const int*)d_in[9];
    const int* batch = (const int*)d_in[10];
    float* out = (float*)d_out;

    const int N = in_sizes[10];          // 100000
    const int K = in_sizes[0] / N;       // 11
    const int E = in_sizes[9] / 2;       // 1600000
    const int G = out_size;              // 4096
    const int* row = edge;
    const int* col = edge + E;

    // workspace layout (floats)
    float* ws   = (float*)d_ws;
    float* dis  = ws;                                    // N
    float* bufA = ws + (((size_t)N + 255) & ~(size_t)255);   // N*128  (xw / pooled / mlp-t)
    float* bufB = bufA + (size_t)N * HID;                // N*128  (h)
    float* pooled = bufA;                                // G*128 (reused after scatter2)
    float* cnt    = bufA + (size_t)G * HID;              // G
    float* tbuf   = bufA + (size_t)G * HID + G;          // G*128

    const long long NH = (long long)N * HID;
    const long long items = (long long)E + N;
    const int mFull = N / 64;            // guard-free 64-row blocks
    const int mTail = N - mFull * 64;    // remaining rows (single guarded block)

    // 1) degree with self-loop, then deg^-1/2
    fill_value_kernel<<<(unsigned)((N + 255) / 256), 256, 0, stream>>>(dis, 1.0f, N);
    deg_accum_kernel<<<(unsigned)((E + 255) / 256), 256, 0, stream>>>(dis, col, E);
    rsqrt_inplace_kernel<<<(unsigned)((N + 255) / 256), 256, 0, stream>>>(dis, N);

    // 2) layer 1: xw = x @ W1 ; h = b1 ; scatter
    gemm_k11_kernel<<<2048, 128, 0, stream>>>(x, W1, bufA, N, K);
    fill_rows_bias_kernel<<<(unsigned)((NH + 255) / 256), 256, 0, stream>>>(bufB, b1, NH);
    scatter_kernel<<<(unsigned)((items * 32 + 255) / 256), 256, 0, stream>>>(
        bufA, bufB, dis, row, col, E, N);

    // 3) layer 2: xw2 = relu(h1) @ W2 (WMMA) ; h2 = b2 ; scatter
    if (mFull > 0)
        wmma_gemm_k128_kernel<true, false, false, false><<<(unsigned)mFull, 256, 0, stream>>>(
            bufB, W2, bufA, nullptr, N, 0);
    if (mTail > 0)
        wmma_gemm_k128_kernel<true, false, false, true><<<1, 256, 0, stream>>>(
            bufB, W2, bufA, nullptr, N, mFull * 64);
    fill_rows_bias_kernel<<<(unsigned)((NH + 255) / 256), 256, 0, stream>>>(bufB, b2, NH);
    scatter_kernel<<<(unsigned)((items * 32 + 255) / 256), 256, 0, stream>>>(
        bufA, bufB, dis, row, col, E, N);

    // 4) global mean pool of relu(h2)
    const long long GP = (long long)G * HID + G;
    fill_value_kernel<<<(unsigned)((GP + 255) / 256), 256, 0, stream>>>(pooled, 0.0f, GP);
    pool_accum_kernel<<<(unsigned)(((long long)N * 32 + 255) / 256), 256, 0, stream>>>(
        bufB, batch, pooled, cnt, N);
    pool_scale_kernel<<<(unsigned)((G * HID + 255) / 256), 256, 0, stream>>>(pooled, cnt, G * HID);

    // 5) MLP: t = relu(pooled @ Wl1 + bl1) (WMMA, G % 64 == 0) ; out = t @ Wl2 + bl2
    const int gFull = G / 64;
    const int gTail = G - gFull * 64;
    if (gFull > 0)
        wmma_gemm_k128_kernel<false, true, true, false><<<(unsigned)gFull, 256, 0, stream>>>(
            pooled, Wl1, tbuf, bl1, G, 0);
    if (gTail > 0)
        wmma_gemm_k128_kernel<false, true, true, true><<<1, 256, 0, stream>>>(
            pooled, Wl1, tbuf, bl1, G, gFull * 64);
    final_dot_kernel<<<(unsigned)(((long long)G * 32 + 255) / 256), 256, 0, stream>>>(
        tbuf, Wl2, bl2, out, G);
}